// fast_rcnn_net_36412732736182
// MI455X (gfx1250) — compile-verified
//
#include <hip/hip_runtime.h>
#include <math.h>

typedef __attribute__((ext_vector_type(2))) float v2f;
typedef __attribute__((ext_vector_type(8))) float v8f;

#define N_ROI   128
#define C_CH    512
#define HF      38
#define WF      38
#define ROI     7
#define KPOOL   (C_CH * ROI * ROI)   // 25088
#define NHID    4096

// ---------------------------------------------------------------------------
// Kernel 1: ROI crop + adaptive max pool (7x7). One thread per (n, c, bin).
// ---------------------------------------------------------------------------
__global__ __launch_bounds__(256)
void roi_pool_kernel(const float* __restrict__ x,
                     const int* __restrict__ regions,
                     float* __restrict__ pooled) {
    const int total = N_ROI * C_CH * 49;
    int idx = blockIdx.x * blockDim.x + threadIdx.x;
    if (idx >= total) return;
    const int bin = idx % 49;
    const int c   = (idx / 49) % C_CH;
    const int n   = idx / (49 * C_CH);
    const int i = bin / ROI;      // pooled row
    const int j = bin % ROI;      // pooled col
    const int r    = regions[n * 4 + 0];  // crop col offset (b[0])
    const int cofs = regions[n * 4 + 1];  // crop row offset (b[1])
    const int w    = regions[n * 4 + 2];
    const int h    = regions[n * 4 + 3];
    const int rs = (i * h) / ROI;
    const int re = ((i + 1) * h + ROI - 1) / ROI;
    const int cs = (j * w) / ROI;
    const int ce = ((j + 1) * w + ROI - 1) / ROI;
    const float* base = x + (((size_t)n * C_CH + c) * HF + cofs) * WF + r;
    float m = -INFINITY;
    for (int y = rs; y < re; ++y) {
        const float* row = base + y * WF;
        for (int xx = cs; xx < ce; ++xx)
            m = fmaxf(m, row[xx]);
    }
    pooled[(size_t)n * KPOOL + c * 49 + bin] = m;
}

// ---------------------------------------------------------------------------
// Kernel 2: C[128,N] = relu(A[128,K] @ B[K,N] + bias), fp32 WMMA 16x16x4.
// Block = 256 threads (8 waves). Block tile = 128 rows x 64 cols.
// Wave wv owns rows [16*wv, 16*wv+16); 4 accumulators cover the 64 cols.
// A chunk:  [row][k]           stride 34 (conflict-free for fragment reads)
// B chunk:  [kpair][n][2]      K-paired so each B fragment is one ds b64 load;
//                              pair-stride 160 puts the two half-waves on
//                              disjoint bank halves (0-31 vs 32-63).
// ---------------------------------------------------------------------------
#define KC       32
#define TN       64
#define ASTRIDE  (KC + 2)        // 34
#define BPSTRIDE (2 * TN + 32)   // 160 floats per k-pair row

__global__ __launch_bounds__(256)
void gemm_bias_relu_f32_wmma(const float* __restrict__ A,
                             const float* __restrict__ B,
                             const float* __restrict__ bias,
                             float* __restrict__ Cout,
                             int K, int N, int do_relu) {
    __shared__ float As[128 * ASTRIDE];
    __shared__ float Bs[(KC / 2) * BPSTRIDE];

    const int tid  = threadIdx.x;
    const int lane = tid & 31;
    const int wv   = tid >> 5;              // wave 0..7 -> M strip
    const int n0   = blockIdx.x * TN;
    const int mrow = wv * 16 + (lane & 15); // A-fragment row (m = lane%16)
    const int kb   = (lane >> 4) * 2;       // k-pair select (lanes 16-31 -> k+2)
    const int ncol = lane & 15;             // B/C/D fragment column

    v8f acc[4] = {};

    for (int k0 = 0; k0 < K; k0 += KC) {
        // Stage A chunk: 128 x 32, contiguous along k (coalesced).
        for (int e = tid; e < 128 * KC; e += 256) {
            int r  = e >> 5;
            int kk = e & (KC - 1);
            As[r * ASTRIDE + kk] = A[(size_t)r * K + k0 + kk];
        }
        // Stage B chunk: 32 x 64 rows coalesced from global, K-paired in LDS.
        for (int e = tid; e < KC * TN; e += 256) {
            int kk = e >> 6;           // k within chunk
            int nn = e & (TN - 1);     // column within tile
            Bs[(kk >> 1) * BPSTRIDE + nn * 2 + (kk & 1)] =
                B[(size_t)(k0 + kk) * N + n0 + nn];
        }
        if (k0 + KC < K)
            __builtin_prefetch(&B[(size_t)(k0 + KC) * N + n0 + (tid & (TN - 1))], 0, 0);
        __syncthreads();

        #pragma unroll
        for (int kk = 0; kk < KC; kk += 4) {
            v2f a;
            a.x = As[mrow * ASTRIDE + kk + kb];
            a.y = As[mrow * ASTRIDE + kk + kb + 1];
            const float* brow = &Bs[((kk + kb) >> 1) * BPSTRIDE];
            #pragma unroll
            for (int t = 0; t < 4; ++t) {
                v2f b = *(const v2f*)&brow[(t * 16 + ncol) * 2]; // one b64 load
                acc[t] = __builtin_amdgcn_wmma_f32_16x16x4_f32(
                    /*neg_a=*/false, a, /*neg_b=*/false, b,
                    /*c_mod=*/(short)0, acc[t],
                    /*reuse_a=*/false, /*reuse_b=*/false);
            }
        }
        __syncthreads();
    }

    // D layout: VGPR v holds row (v + 8*(lane/16)) of the wave tile, col = lane%16.
    const int rbase = wv * 16 + (lane >> 4) * 8;
    #pragma unroll
    for (int t = 0; t < 4; ++t) {
        const int ng = n0 + t * 16 + ncol;
        const float bv = bias[ng];
        #pragma unroll
        for (int v = 0; v < 8; ++v) {
            float val = acc[t][v] + bv;
            if (do_relu) val = val > 0.0f ? val : 0.0f;
            Cout[(size_t)(rbase + v) * N + ng] = val;
        }
    }
}

// ---------------------------------------------------------------------------
// Kernel 3: heads. One block per ROI: h2 row in LDS, 21+80 dot products,
// softmax over 21 classes, ceil/floor/exp bbox transform.
// ---------------------------------------------------------------------------
__global__ __launch_bounds__(128)
void heads_kernel(const float* __restrict__ H2,
                  const float* __restrict__ Wcls,
                  const float* __restrict__ Wbox,
                  const int* __restrict__ regions,
                  float* __restrict__ out_clf,
                  float* __restrict__ out_box) {
    __shared__ float hrow[NHID];
    __shared__ float scores[21];
    __shared__ float tvals[80];
    __shared__ float inv_sum;

    const int n = blockIdx.x;
    const int t = threadIdx.x;

    for (int k = t; k < NHID; k += 128)
        hrow[k] = H2[(size_t)n * NHID + k];
    __syncthreads();

    if (t < 21) {
        float s = 0.0f;
        for (int k = 0; k < NHID; ++k) s = fmaf(hrow[k], Wcls[k * 21 + t], s);
        scores[t] = s;
    } else if (t < 101) {
        const int j = t - 21;
        float s = 0.0f;
        for (int k = 0; k < NHID; ++k) s = fmaf(hrow[k], Wbox[k * 80 + j], s);
        tvals[j] = s;
    }
    __syncthreads();

    if (t == 0) {
        float mx = scores[0];
        for (int i = 1; i < 21; ++i) mx = fmaxf(mx, scores[i]);
        float sum = 0.0f;
        for (int i = 0; i < 21; ++i) { float e = expf(scores[i] - mx); scores[i] = e; sum += e; }
        inv_sum = 1.0f / sum;
    }
    __syncthreads();

    if (t < 21)
        out_clf[n * 21 + t] = scores[t] * inv_sum;

    if (t < 20) {
        const float r = (float)regions[n * 4 + 0];
        const float c = (float)regions[n * 4 + 1];
        const float w = (float)regions[n * 4 + 2];
        const float h = (float)regions[n * 4 + 3];
        const float t0 = tvals[4 * t + 0];
        const float t1 = tvals[4 * t + 1];
        const float t2 = tvals[4 * t + 2];
        const float t3 = tvals[4 * t + 3];
        const float S = 16.0f;  // STRIDE
        out_box[n * 80 + 4 * t + 0] = ceilf((w * t0 + r) * S) - 1.0f;
        out_box[n * 80 + 4 * t + 1] = ceilf((h * t1 + c) * S) - 1.0f;
        out_box[n * 80 + 4 * t + 2] = floorf(w * expf(t2) * S) + 1.0f;
        out_box[n * 80 + 4 * t + 3] = floorf(h * expf(t3) * S) + 1.0f;
    }
}

// ---------------------------------------------------------------------------
extern "C" void kernel_launch(void* const* d_in, const int* in_sizes, int n_in,
                              void* d_out, int out_size, void* d_ws, size_t ws_size,
                              hipStream_t stream) {
    const float* x       = (const float*)d_in[0];
    const int*   regions = (const int*)  d_in[1];
    const float* W1      = (const float*)d_in[2];
    const float* b1      = (const float*)d_in[3];
    const float* W2      = (const float*)d_in[4];
    const float* b2      = (const float*)d_in[5];
    const float* Wcls    = (const float*)d_in[6];
    const float* Wbox    = (const float*)d_in[7];
    float* out = (float*)d_out;

    float* pooled = (float*)d_ws;                         // 128 x 25088
    float* h1     = pooled + (size_t)N_ROI * KPOOL;       // 128 x 4096
    float* h2     = h1 + (size_t)N_ROI * NHID;            // 128 x 4096

    {
        const int total = N_ROI * C_CH * 49;
        roi_pool_kernel<<<(total + 255) / 256, 256, 0, stream>>>(x, regions, pooled);
    }
    // h1 = relu(pooled @ W1 + b1)   K=25088 (784 chunks of 32), N=4096, 64 blocks
    gemm_bias_relu_f32_wmma<<<NHID / TN, 256, 0, stream>>>(pooled, W1, b1, h1,
                                                           KPOOL, NHID, 1);
    // h2 = relu(h1 @ W2 + b2)       K=4096, N=4096
    gemm_bias_relu_f32_wmma<<<NHID / TN, 256, 0, stream>>>(h1, W2, b2, h2,
                                                           NHID, NHID, 1);
    // heads: softmax(h2@Wcls), bbox transform of h2@Wbox
    heads_kernel<<<N_ROI, 128, 0, stream>>>(h2, Wcls, Wbox, regions,
                                            out, out + N_ROI * 21);
}